// MHLocalDenseSynthesizerAttention_3770981286600
// MI455X (gfx1250) — compile-verified
//
#include <hip/hip_runtime.h>

typedef float v2f __attribute__((ext_vector_type(2)));
typedef float v4f __attribute__((ext_vector_type(4)));
typedef float v8f __attribute__((ext_vector_type(8)));
typedef unsigned int v4u __attribute__((ext_vector_type(4)));
typedef int v4i __attribute__((ext_vector_type(4)));
typedef int v8i __attribute__((ext_vector_type(8)));

#if defined(__has_builtin)
#if __has_builtin(__builtin_amdgcn_tensor_load_to_lds) && \
    __has_builtin(__builtin_amdgcn_s_wait_tensorcnt)
#define USE_TDM 1
#endif
#endif

// ---------------------------------------------------------------------------
// GEMM: out[m,n] = act( sum_k in[m,k] * W[n,k] ),  in:[M,K] row-major,
// W:[N,K] row-major (i.e. out = in @ W^T).  M=16384, N=512, K=512.
// Native fp32 matrix path: V_WMMA_F32_16X16X4_F32 (matches reference dtype).
// Block: 256 thr (8 waves). Tile: 128 rows x 64 cols. Wave tile: 16x64.
// Weight slice staged into LDS by the Tensor Data Mover (one TDM descriptor
// per 64x128 K-chunk, issued by wave 0, tracked by TENSORcnt). TDM's LDS pad
// feature (4 DWORDs after every 128 DWORDs) reproduces the 132-float padded
// row stride that keeps ds_load_2addr_b64 bank-conflict-free.
// ---------------------------------------------------------------------------
template <bool RELU>
__global__ __launch_bounds__(256) void gemm_wmma_f32(
    const float* __restrict__ in, const float* __restrict__ W,
    float* __restrict__ out, int M, int N, int K) {
  constexpr int KB  = 128;     // K-chunk staged in LDS
  constexpr int LDB = KB + 4;  // padded LDS row stride (floats)
  __shared__ float Bs[64 * LDB];

  const int m0  = blockIdx.x * 128;
  const int n0  = blockIdx.y * 64;
  const int tid = threadIdx.x;
  const int wid  = tid >> 5;            // wave id 0..7 -> row sub-tile
  const int lane = tid & 31;
  const int lr   = lane & 15;           // row/col within fragment
  const int kh   = (lane >> 4) << 1;    // 0 or 2: K sub-offset per half-wave

  const float* arow = in + (size_t)(m0 + wid * 16 + lr) * K + kh;

  v8f acc0 = {}, acc1 = {}, acc2 = {}, acc3 = {};

  for (int kb = 0; kb < K; kb += KB) {
#ifdef USE_TDM
    // ---- TDM: DMA W[n0..n0+63][kb..kb+KB-1] -> LDS (padded rows) ----
    if (wid == 0) {
      const unsigned long long ga =
          (unsigned long long)(uintptr_t)(W + (size_t)n0 * K + kb);
      const unsigned int lds_base = (unsigned int)(uintptr_t)(&Bs[0]);
      // D# group 0: count=1 | lds_addr | global_addr[56:0] | type=2
      const v4u g0 = {1u, lds_base, (unsigned int)(ga & 0xFFFFFFFFull),
                      (unsigned int)((ga >> 32) & 0x01FFFFFFull) | (2u << 30)};
      // D# group 1: data_size=4B, pad_enable, pad_interval=6 (128 DW),
      // pad_amount=3 (4 DW); tensor_dim0=K, tensor_dim1=512 rows;
      // tile_dim0=KB, tile_dim1=64; tensor_dim0_stride=K.
      const v8i g1 = {
          (int)((2u << 16) | (1u << 20) | (6u << 22) | (3u << 25)),
          (int)((unsigned int)K << 16),    // tensor_dim0[15:0] @ bits 63:48
          (int)(512u << 16),               // tensor_dim1[15:0] @ bits 95:80
          (int)((unsigned int)KB << 16),   // tile_dim0 @ bits 127:112
          64,                              // tile_dim1 @ bits 143:128
          K,                               // tensor_dim0_stride[31:0]
          0, 0};
      const v4i gz = {0, 0, 0, 0};         // groups 2/3 unused (2-D tensor)
#if __clang_major__ >= 23
      const v8i gz8 = {0, 0, 0, 0, 0, 0, 0, 0};
      __builtin_amdgcn_tensor_load_to_lds(g0, g1, gz, gz, gz8, 0);
#else
      __builtin_amdgcn_tensor_load_to_lds(g0, g1, gz, gz, 0);
#endif
      __builtin_amdgcn_s_wait_tensorcnt(0);
    }
    __syncthreads();
#else
    // ---- fallback: stage via float4 copy loop ----
    for (int i = tid; i < 64 * (KB / 4); i += 256) {
      const int n  = i >> 5;  // KB/4 == 32 float4 per row
      const int k4 = i & 31;
      v4f w = *(const v4f*)(W + (size_t)(n0 + n) * K + kb + k4 * 4);
      *(v4f*)(&Bs[n * LDB + k4 * 4]) = w;
    }
    __syncthreads();
#endif

    // ---- K inner loop: 1 A-frag b64 load feeds 4 WMMAs ----
#pragma unroll 4
    for (int k0 = 0; k0 < KB; k0 += 4) {
      v2f a  = *(const v2f*)(arow + kb + k0);
      v2f b0 = *(const v2f*)(&Bs[(0 * 16 + lr) * LDB + k0 + kh]);
      v2f b1 = *(const v2f*)(&Bs[(1 * 16 + lr) * LDB + k0 + kh]);
      v2f b2 = *(const v2f*)(&Bs[(2 * 16 + lr) * LDB + k0 + kh]);
      v2f b3 = *(const v2f*)(&Bs[(3 * 16 + lr) * LDB + k0 + kh]);
      acc0 = __builtin_amdgcn_wmma_f32_16x16x4_f32(false, a, false, b0,
                                                   (short)0, acc0, false, false);
      acc1 = __builtin_amdgcn_wmma_f32_16x16x4_f32(false, a, false, b1,
                                                   (short)0, acc1, false, false);
      acc2 = __builtin_amdgcn_wmma_f32_16x16x4_f32(false, a, false, b2,
                                                   (short)0, acc2, false, false);
      acc3 = __builtin_amdgcn_wmma_f32_16x16x4_f32(false, a, false, b3,
                                                   (short)0, acc3, false, false);
    }
    __syncthreads();
  }

  // ---- store: C/D layout => VGPR j holds row j (lanes 0-15) / j+8 (16-31),
  //      col = n0 + tile*16 + (lane&15) ----
  const int rowbase = m0 + wid * 16 + ((lane >> 4) << 3);
#pragma unroll
  for (int j = 0; j < 8; ++j) {
    float r0 = acc0[j], r1 = acc1[j], r2 = acc2[j], r3 = acc3[j];
    if (RELU) {
      r0 = fmaxf(r0, 0.f); r1 = fmaxf(r1, 0.f);
      r2 = fmaxf(r2, 0.f); r3 = fmaxf(r3, 0.f);
    }
    const size_t rb = (size_t)(rowbase + j) * N + n0 + lr;
    out[rb +  0] = r0;
    out[rb + 16] = r1;
    out[rb + 32] = r2;
    out[rb + 48] = r3;
  }
}

// ---------------------------------------------------------------------------
// attn[bt,h,c] = softmax_c( sum_d relu_q[bt, h*64+d] * w2[h*15+c, d] + b2 )
// One thread per (bt, h). w2 kept transposed in LDS: w2s[d][h*15+c] so that
// lanes (varying h) hit distinct banks; same-h lanes broadcast.
// ---------------------------------------------------------------------------
__global__ __launch_bounds__(256) void attn_kernel(
    const float* __restrict__ q, const float* __restrict__ w2,
    const float* __restrict__ b2, float* __restrict__ attn) {
  __shared__ float w2s[64 * 120];  // [d][h*15+c]
  __shared__ float b2s[120];
  for (int i = threadIdx.x; i < 120 * 64; i += 256) {
    const int r = i >> 6;  // 0..119  (h*15+c)
    const int d = i & 63;
    w2s[d * 120 + r] = w2[i];
  }
  for (int i = threadIdx.x; i < 120; i += 256) b2s[i] = b2[i];
  __syncthreads();

  const int gid = blockIdx.x * 256 + threadIdx.x;  // bt*8 + h, 131072 total
  const int h   = gid & 7;
  const int bt  = gid >> 3;
  const float* qp = q + (size_t)bt * 512 + h * 64;

  float logits[15];
#pragma unroll
  for (int c = 0; c < 15; ++c) logits[c] = b2s[h * 15 + c];

#pragma unroll 4
  for (int d4 = 0; d4 < 16; ++d4) {
    v4f qa = *(const v4f*)(qp + d4 * 4);
#pragma unroll
    for (int e = 0; e < 4; ++e) {
      const float qv = qa[e];
      const int   d  = d4 * 4 + e;
#pragma unroll
      for (int c = 0; c < 15; ++c)
        logits[c] = fmaf(qv, w2s[d * 120 + h * 15 + c], logits[c]);
    }
  }

  float m = logits[0];
#pragma unroll
  for (int c = 1; c < 15; ++c) m = fmaxf(m, logits[c]);
  float s = 0.f;
#pragma unroll
  for (int c = 0; c < 15; ++c) {
    logits[c] = __expf(logits[c] - m);
    s += logits[c];
  }
  const float inv = 1.f / s;
  float* ap = attn + (size_t)gid * 15;
#pragma unroll
  for (int c = 0; c < 15; ++c) ap[c] = logits[c] * inv;
}

// ---------------------------------------------------------------------------
// x[bt, col] = sum_{c=0..14} attn[bt, h(col), c] * v[bt + c - 7, col]
// (taps outside the batch's [0,2048) t-range contribute zero).
// One float4 per thread; v re-reads are L2 hits (v = 32MB << 192MB L2).
// ---------------------------------------------------------------------------
__global__ __launch_bounds__(256) void mix_kernel(
    const float* __restrict__ v, const float* __restrict__ attn,
    float* __restrict__ x) {
  const int gid  = blockIdx.x * 256 + threadIdx.x;  // 16384*512/4 items
  const int col4 = gid & 127;
  const int bt   = gid >> 7;
  const int t    = bt & 2047;
  const int h    = col4 >> 4;

  const float* ap = attn + ((size_t)bt * 8 + h) * 15;
  v4f acc = {0.f, 0.f, 0.f, 0.f};
#pragma unroll
  for (int c = 0; c < 15; ++c) {
    const int tt = t + c - 7;
    if (tt >= 0 && tt < 2048) {
      const float a = ap[c];
      v4f vv = *(const v4f*)(v + (size_t)(bt + c - 7) * 512 + col4 * 4);
      acc += a * vv;
    }
  }
  *(v4f*)(x + (size_t)bt * 512 + col4 * 4) = acc;
}

// ---------------------------------------------------------------------------
extern "C" void kernel_launch(void* const* d_in, const int* in_sizes, int n_in,
                              void* d_out, int out_size, void* d_ws,
                              size_t ws_size, hipStream_t stream) {
  const float* query = (const float*)d_in[0];
  const float* key_u = (const float*)d_in[1];  // unused by reference math
  const float* value = (const float*)d_in[2];
  // d_in[3] = mask (dummy, unused)
  const float* w1 = (const float*)d_in[4];
  const float* w2 = (const float*)d_in[5];
  const float* b2 = (const float*)d_in[6];
  const float* w3 = (const float*)d_in[7];
  const float* wo = (const float*)d_in[8];
  (void)key_u; (void)in_sizes; (void)n_in; (void)out_size; (void)ws_size;

  float* out = (float*)d_out;

  const int M = 16384, N = 512, K = 512;
  float* ws_q    = (float*)d_ws;          // 16384*512 f32 (q, later x)
  float* ws_attn = ws_q + (size_t)M * N;  // 16384*8*15 f32

  dim3 gg(M / 128, N / 64);  // 128 x 8 blocks

  // 1) q = relu(query @ w1^T)            -> ws_q
  gemm_wmma_f32<true><<<gg, 256, 0, stream>>>(query, w1, ws_q, M, N, K);
  // 2) attn = softmax(per-head q @ w2^T + b2)  -> ws_attn
  attn_kernel<<<(M * 8) / 256, 256, 0, stream>>>(ws_q, w2, b2, ws_attn);
  // 3) v = value @ w3^T                  -> d_out (used as scratch here)
  gemm_wmma_f32<false><<<gg, 256, 0, stream>>>(value, w3, out, M, N, K);
  // 4) x = local 15-tap mix(attn, v)     -> ws_q (q is dead now)
  mix_kernel<<<(M * N / 4) / 256, 256, 0, stream>>>(out, ws_attn, ws_q);
  // 5) out = x @ w_out^T                 -> d_out
  gemm_wmma_f32<false><<<gg, 256, 0, stream>>>(ws_q, wo, out, M, N, K);
}